// fdm_refine_block_30313879175572
// MI455X (gfx1250) — compile-verified
//
#include <hip/hip_runtime.h>
#include <hip/hip_bf16.h>
#include <math.h>
#include <stdint.h>

typedef float v2f __attribute__((ext_vector_type(2)));
typedef float v8f __attribute__((ext_vector_type(8)));

#define EPSV   1e-5f
#define HW     300
#define PW     302
#define PLANE  (PW*PW)     // 91204 padded plane
#define OPLANE (HW*HW)     // 90000 unpadded plane

// ---------------------------------------------------------------- WMMA helper
__device__ __forceinline__ v8f wmma_f32x4(v2f a, v2f b, v8f c) {
  // V_WMMA_F32_16X16X4_F32 : D(16x16,f32) = A(16x4,f32) x B(4x16,f32) + C
  return __builtin_amdgcn_wmma_f32_16x16x4_f32(false, a, false, b, (short)0, c,
                                               false, false);
}

// ----------------------------------------------- async global->LDS (CDNA5)
__device__ __forceinline__ void async_load_lds_b32(const float* gaddr,
                                                   float* ldsaddr) {
  uint32_t lo = (uint32_t)(uintptr_t)ldsaddr;      // flat low 32 = LDS offset
  asm volatile("global_load_async_to_lds_b32 %0, %1, off"
               :: "v"(lo), "v"((uint64_t)(uintptr_t)gaddr)
               : "memory");
}
__device__ __forceinline__ void wait_asynccnt0() {
#if __has_builtin(__builtin_amdgcn_s_wait_asynccnt)
  __builtin_amdgcn_s_wait_asynccnt(0);
#else
  asm volatile("s_wait_asynccnt 0x0" ::: "memory");
#endif
}

// ------------------------------------------------- bilinear upsample (2 taps)
__device__ __forceinline__ float bilin_up(const float* __restrict__ img, int s,
                                          int oh, int ow) {
  float scale = (float)s / (float)HW;          // in/out < 1 -> upsample
  float sy = (oh + 0.5f) * scale - 0.5f;
  float sx = (ow + 0.5f) * scale - 0.5f;
  int y0 = (int)floorf(sy); float fy = sy - (float)y0;
  int x0 = (int)floorf(sx); float fx = sx - (float)x0;
  int y0c = min(max(y0, 0), s-1), y1c = min(max(y0+1, 0), s-1);
  int x0c = min(max(x0, 0), s-1), x1c = min(max(x0+1, 0), s-1);
  float a = img[y0c*s + x0c], b = img[y0c*s + x1c];
  float c = img[y1c*s + x0c], d = img[y1c*s + x1c];
  return (1.f-fy)*((1.f-fx)*a + fx*b) + fy*((1.f-fx)*c + fx*d);
}

// ----------------------------------- zero GN/bcg stats + base_pad halo rings
__global__ void init_kernel(float* __restrict__ basePad, float* __restrict__ stats) {
  int idx = blockIdx.x*256 + threadIdx.x;
  if (idx < 32) stats[idx] = 0.f;
  if (idx >= 4*64*1204) return;
  int p = idx / 1204; int r = idx - p*1204;
  int row, col;
  if (r < 302)      { row = 0;   col = r; }
  else if (r < 604) { row = 301; col = r - 302; }
  else { int rr = r - 604; row = 1 + (rr >> 1); col = (rr & 1) ? 301 : 0; }
  basePad[(size_t)p*PLANE + row*PW + col] = 0.f;
}

// -------------------------- weights OIHW -> k-major (tap,cin)-by-m for GEMM A
__global__ void wt_transform(const float* __restrict__ W, float* __restrict__ Wt,
                             int CIN) {
  int idx = blockIdx.x*256 + threadIdx.x;
  if (idx >= 9*CIN*64) return;
  int m = idx & 63; int k = idx >> 6; int c = k % CIN; int tap = k / CIN;
  Wt[idx] = W[((size_t)m*CIN + c)*9 + tap];
}

// ------------- upsample x2/x3/x4 -> padded concat (192,302,302), one batch b
__global__ void resize_concat_1b(const float* __restrict__ x2,
                                 const float* __restrict__ x3,
                                 const float* __restrict__ x4,
                                 float* __restrict__ xcat, int b) {
  int idx = blockIdx.x*256 + threadIdx.x;     // < 192*91204 (exact grid)
  int c = idx / PLANE;
  int pix = idx - c*PLANE;
  int pr = pix / PW, pc = pix - pr*PW;
  float v = 0.f;
  if (pr >= 1 && pr <= HW && pc >= 1 && pc <= HW) {
    const float* src; int s; int cc;
    if (c < 64)       { src = x2; s = 75; cc = c; }
    else if (c < 128) { src = x3; s = 38; cc = c - 64; }
    else              { src = x4; s = 19; cc = c - 128; }
    v = bilin_up(src + (size_t)(b*64 + cc)*s*s, s, pr-1, pc-1);
  }
  xcat[idx] = v;
}

// ---------------------------------------------------------------------------
// 3x3 conv as implicit GEMM with V_WMMA_F32_16X16X4_F32.
// Y[m][oh][ow] = sum_{tap,c} W[m][c][tap] * Xpad[c][oh+kh][ow+kw] + bias[m]
// Block = 256 thr (8 waves). Tile: M=64 (all out ch), N=128 cols of one row.
// Wave w -> m-tile (w&3), n-group (w>>2) of 4 n-tiles. K chunks of 4 channels.
// A frag (16x4): lane L: M=L%16, K = v + 2*(L/16)  (ISA 7.12.2)
// B frag (4x16): lane L: N=L%16, K = v + 2*(L/16)
// D      (16x16): vgpr r: M = r + 8*(L/16), N = L%16
// Input tile staged with GLOBAL_LOAD_ASYNC_TO_LDS_B32 (ASYNCcnt).
// ---------------------------------------------------------------------------
template<int CIN>
__global__ __launch_bounds__(256)
void conv3x3_wmma(const float* __restrict__ Xpad,   // (CIN,302,302) one sample
                  const float* __restrict__ Wt,     // (9*CIN, 64) k-major
                  const float* __restrict__ bias,   // (64)
                  float* __restrict__ Y, int outPlane, int outRow, int outOff) {
  __shared__ float Wlds[9][16][64];                 // 36.9 KB
  __shared__ float Xlds[16][3][132];                // 25.3 KB
  const int tid = threadIdx.x;
  const int wv = tid >> 5, lane = tid & 31, laneM = lane & 15, half = lane >> 4;
  const int oh  = blockIdx.y;
  const int cw0 = blockIdx.x * 128;
  const int nvalid = min(128, HW - cw0);
  const int mtile = wv & 3, ngrp = wv >> 2;
  const bool ragged = (cw0 + 129 >= PW);            // right-edge tile

  v8f acc[4];
  #pragma unroll
  for (int i = 0; i < 4; ++i)
    #pragma unroll
    for (int r = 0; r < 8; ++r) acc[i][r] = 0.f;

  for (int c0 = 0; c0 < CIN; c0 += 16) {
    __syncthreads();
    // stage weights for this channel chunk: [tap][c'][m]
    for (int t = tid; t < 9*16*64; t += 256) {
      int m = t & 63, tc = t >> 6, cc = tc & 15, tap = tc >> 4;
      Wlds[tap][cc][m] = Wt[(size_t)(tap*CIN + c0 + cc)*64 + m];
    }
    // stage input tile (16 ch x 3 rows x 130 cols) via async global->LDS
    for (int t = tid; t < 16*3*130; t += 256) {
      int col = t % 130, tr = t / 130, dr = tr % 3, ch = tr / 3;
      int pcol = min(cw0 + col, PW - 1);            // clamp; OOB fixed below
      async_load_lds_b32(&Xpad[(size_t)(c0 + ch)*PLANE + (oh + dr)*PW + pcol],
                         &Xlds[ch][dr][col]);
    }
    // prefetch next channel chunk (global_prefetch_b8)
    if (c0 + 16 < CIN) {
      int ch = tid & 15, dr = (tid >> 4) & 3, cb = tid >> 6;
      int pcol = cw0 + cb*34;
      if (dr < 3 && pcol < PW)
        __builtin_prefetch(&Xpad[(size_t)(c0 + 16 + ch)*PLANE + (oh + dr)*PW + pcol], 0, 1);
    }
    wait_asynccnt0();
    if (ragged) {                                   // zero-fill past right edge
      for (int t = tid; t < 16*3*130; t += 256) {
        int col = t % 130, tr = t / 130, dr = tr % 3, ch = tr / 3;
        if (cw0 + col >= PW) Xlds[ch][dr][col] = 0.f;
      }
    }
    __syncthreads();

    for (int tap = 0; tap < 9; ++tap) {
      const int kh = tap / 3, kw = tap - kh*3;
      #pragma unroll
      for (int kc = 0; kc < 4; ++kc) {
        v2f a;
        a[0] = Wlds[tap][kc*4 + 2*half + 0][mtile*16 + laneM];
        a[1] = Wlds[tap][kc*4 + 2*half + 1][mtile*16 + laneM];
        #pragma unroll
        for (int nt = 0; nt < 4; ++nt) {
          int colb = (ngrp*4 + nt)*16 + laneM + kw;
          v2f bf;
          bf[0] = Xlds[kc*4 + 2*half + 0][kh][colb];
          bf[1] = Xlds[kc*4 + 2*half + 1][kh][colb];
          acc[nt] = wmma_f32x4(a, bf, acc[nt]);
        }
      }
    }
  }

  #pragma unroll
  for (int nt = 0; nt < 4; ++nt) {
    int col = (ngrp*4 + nt)*16 + laneM;
    if (col < nvalid) {
      #pragma unroll
      for (int r = 0; r < 8; ++r) {
        int m = mtile*16 + r + 8*half;
        Y[(size_t)m*outPlane + oh*outRow + (cw0 + col) + outOff] = acc[nt][r] + bias[m];
      }
    }
  }
}

// ----------------------------------------- GroupNorm(1) stats for one sample
__global__ void gn_stats_1(const float* __restrict__ X, int plane, int rowStride,
                           int off, float* __restrict__ stat2) {
  const int tid = threadIdx.x;
  float s1 = 0.f, s2 = 0.f;
  for (int i = blockIdx.x*blockDim.x + tid; i < 64*OPLANE;
       i += gridDim.x*blockDim.x) {
    int c = i / OPLANE; int rem = i - c*OPLANE;
    int oh = rem / HW, ow = rem - oh*HW;
    float x = X[(size_t)c*plane + oh*rowStride + ow + off];
    s1 += x; s2 += x*x;
  }
  __shared__ float r1[256], r2[256];
  r1[tid] = s1; r2[tid] = s2; __syncthreads();
  for (int o = 128; o > 0; o >>= 1) {
    if (tid < o) { r1[tid] += r1[tid+o]; r2[tid] += r2[tid+o]; }
    __syncthreads();
  }
  if (tid == 0) { atomicAdd(&stat2[0], r1[0]); atomicAdd(&stat2[1], r2[0]); }
}

__global__ void gn_norm_relu_1(float* __restrict__ X, int plane, int rowStride,
                               int off, const float* __restrict__ stat2,
                               const float* __restrict__ g,
                               const float* __restrict__ be) {
  int idx = blockIdx.x*256 + threadIdx.x;
  if (idx >= 64*OPLANE) return;
  int c = idx / OPLANE; int rem = idx - c*OPLANE;
  int oh = rem / HW, ow = rem - oh*HW;
  const float invN = 1.f / (64.f * (float)OPLANE);
  float mu  = stat2[0] * invN;
  float var = stat2[1] * invN - mu*mu;
  float inv = rsqrtf(var + EPSV);
  size_t a = (size_t)c*plane + oh*rowStride + ow + off;
  float v = (X[a] - mu) * inv * g[c] + be[c];
  X[a] = v > 0.f ? v : 0.f;
}

// ------------------------------------------ bcg = sigmoid(conv1x1) + its sum
__global__ void bcg_kernel(const float* __restrict__ basePad,
                           const float* __restrict__ Wc,
                           const float* __restrict__ b0,
                           float* __restrict__ bcgOut,
                           float* __restrict__ bcgSum) {
  int b = blockIdx.y;
  int p = blockIdx.x*256 + threadIdx.x;
  float sig = 0.f;
  if (p < OPLANE) {
    int oh = p / HW, ow = p - oh*HW;
    const float* base = basePad + (size_t)b*64*PLANE + (oh+1)*PW + (ow+1);
    float acc = b0[0];
    #pragma unroll 8
    for (int c = 0; c < 64; ++c) acc += base[(size_t)c*PLANE] * Wc[c];
    sig = 1.f / (1.f + expf(-acc));
    bcgOut[(size_t)b*OPLANE + p] = sig;
  }
  __shared__ float rs[256];
  rs[threadIdx.x] = sig; __syncthreads();
  for (int o = 128; o > 0; o >>= 1) {
    if (threadIdx.x < o) rs[threadIdx.x] += rs[threadIdx.x+o];
    __syncthreads();
  }
  if (threadIdx.x == 0) atomicAdd(&bcgSum[b], rs[0]);
}

__global__ void fdm_mask_kernel(const float* __restrict__ fdm,
                                const float* __restrict__ bcg,
                                const float* __restrict__ bcgSum,
                                float* __restrict__ out) {
  int idx = blockIdx.x*256 + threadIdx.x;
  if (idx >= 4*OPLANE) return;
  int b = idx / OPLANE;
  float mean = bcgSum[b] * (1.f / (float)OPLANE);
  out[idx] = (bcg[idx] > mean) ? fdm[idx] : 0.f;
}

// ----------------------------------------------- adaptive max pool -> 30x30
__global__ void pool_kernel(const float* __restrict__ fdmM,
                            float* __restrict__ pooled) {
  int idx = blockIdx.x*256 + threadIdx.x;
  if (idx >= 4*900) return;
  int b = idx / 900; int rem = idx - b*900; int r = rem/30, cc = rem - r*30;
  const float* p = fdmM + (size_t)b*OPLANE + (r*10)*HW + cc*10;
  float m = -3.4e38f;
  for (int dy = 0; dy < 10; ++dy)
    for (int dx = 0; dx < 10; ++dx) m = fmaxf(m, p[dy*HW + dx]);
  pooled[idx] = m;
}

// ---------- antialiased 300->30 downscale (jax.image.resize triangle kernel)
__global__ void resize_w_down_1b(const float* __restrict__ in,   // (64,300,300)
                                 float* __restrict__ out) {      // (64,300,30)
  int idx = blockIdx.x*256 + threadIdx.x;
  if (idx >= 64*HW*30) return;
  int j = idx % 30; int t = idx / 30; int r = t % HW; int c = t / HW;
  float sample = (j + 0.5f)*10.f - 0.5f;
  int k0 = max((int)ceilf(sample - 10.f), 0);
  int k1 = min((int)floorf(sample + 10.f), HW-1);
  const float* row = in + (size_t)c*OPLANE + (size_t)r*HW;
  float s = 0.f, wsum = 0.f;
  for (int k = k0; k <= k1; ++k) {
    float w = fmaxf(1.f - fabsf((float)k - sample)*0.1f, 0.f);
    s += w*row[k]; wsum += w;
  }
  out[idx] = s / wsum;
}

__global__ void resize_h_down_1b(const float* __restrict__ in,   // (64,300,30)
                                 float* __restrict__ out) {      // (64,30,30)
  int idx = blockIdx.x*256 + threadIdx.x;
  if (idx >= 64*900) return;
  int j = idx % 30; int t = idx / 30; int i = t % 30; int c = t / 30;
  float sample = (i + 0.5f)*10.f - 0.5f;
  int k0 = max((int)ceilf(sample - 10.f), 0);
  int k1 = min((int)floorf(sample + 10.f), HW-1);
  const float* col = in + (size_t)c*HW*30 + j;
  float s = 0.f, wsum = 0.f;
  for (int k = k0; k <= k1; ++k) {
    float w = fmaxf(1.f - fabsf((float)k - sample)*0.1f, 0.f);
    s += w*col[k*30]; wsum += w;
  }
  out[idx] = s / wsum;
}

// ------------------------------------------------------------------------
// nl[b,i] = max_j sum_c (pooled[b,i]*nf30[b,c,i]) * nf30[b,c,j] via WMMA f32.
// Block: (itile, b). 8 waves; wave w covers j-tiles w, w+8, ... with a
// running per-lane max; padded j masked with -inf; LDS max reduce at the end.
// ------------------------------------------------------------------------
__global__ __launch_bounds__(256)
void nl_einsum_wmma(const float* __restrict__ nf30,   // (4,64,900)
                    const float* __restrict__ pooled, // (4,900)
                    float* __restrict__ nl) {         // (4,900)
  const int b = blockIdx.y;
  const int it = blockIdx.x;                      // 0..56
  const int tid = threadIdx.x, wv = tid >> 5, lane = tid & 31;
  const int laneM = lane & 15, half = lane >> 4;
  const int i0 = it*16;
  const int iA = i0 + laneM;
  const bool iv = iA < 900;
  const float pl = iv ? pooled[b*900 + iA] : 0.f;
  const float* nfB = nf30 + (size_t)b*64*900;

  v2f af[16];
  #pragma unroll
  for (int kc = 0; kc < 16; ++kc) {
    int k = kc*4 + 2*half;
    af[kc][0] = iv ? pl * nfB[(size_t)k*900 + iA]     : 0.f;
    af[kc][1] = iv ? pl * nfB[(size_t)(k+1)*900 + iA] : 0.f;
  }
  float mx[8];
  #pragma unroll
  for (int r = 0; r < 8; ++r) mx[r] = -3.4e38f;

  for (int jt = wv; jt < 57; jt += 8) {
    int j = jt*16 + laneM;
    bool jv = j < 900;
    int jc = jv ? j : 0;
    v8f acc;
    #pragma unroll
    for (int r = 0; r < 8; ++r) acc[r] = 0.f;
    #pragma unroll
    for (int kc = 0; kc < 16; ++kc) {
      int k = kc*4 + 2*half;
      v2f bf;
      bf[0] = nfB[(size_t)k*900 + jc];
      bf[1] = nfB[(size_t)(k+1)*900 + jc];
      acc = wmma_f32x4(af[kc], bf, acc);
    }
    #pragma unroll
    for (int r = 0; r < 8; ++r)
      mx[r] = fmaxf(mx[r], jv ? acc[r] : -3.4e38f);
  }

  __shared__ float lmax[8][32][8];
  #pragma unroll
  for (int r = 0; r < 8; ++r) lmax[wv][lane][r] = mx[r];
  __syncthreads();
  if (tid < 16) {
    int r = tid & 7, hf = tid >> 3;
    float m = -3.4e38f;
    for (int ww = 0; ww < 8; ++ww)
      for (int l = 0; l < 16; ++l)
        m = fmaxf(m, lmax[ww][hf*16 + l][r]);
    int i = i0 + r + 8*hf;
    if (i < 900) nl[b*900 + i] = m;
  }
}

// ---------- fdm_exp = upsample(nl); base2 = fdm_exp*base; sup = conv1x1(base2)
__global__ void final_kernel(const float* __restrict__ basePad,
                             const float* __restrict__ nl,
                             const float* __restrict__ supW,
                             const float* __restrict__ supB,
                             float* __restrict__ base2,
                             float* __restrict__ supOut) {
  int idx = blockIdx.x*256 + threadIdx.x;
  if (idx >= 4*OPLANE) return;
  int b = idx / OPLANE; int rem = idx - b*OPLANE;
  int oh = rem / HW, ow = rem - oh*HW;
  float sy = (oh + 0.5f)*0.1f - 0.5f;
  float sx = (ow + 0.5f)*0.1f - 0.5f;
  int y0 = (int)floorf(sy); float fy = sy - (float)y0;
  int x0 = (int)floorf(sx); float fx = sx - (float)x0;
  int y0c = min(max(y0, 0), 29), y1c = min(max(y0+1, 0), 29);
  int x0c = min(max(x0, 0), 29), x1c = min(max(x0+1, 0), 29);
  const float* nb = nl + b*900;
  float fe = (1.f-fy)*((1.f-fx)*nb[y0c*30+x0c] + fx*nb[y0c*30+x1c])
           +      fy *((1.f-fx)*nb[y1c*30+x0c] + fx*nb[y1c*30+x1c]);
  const float* base = basePad + (size_t)b*64*PLANE + (oh+1)*PW + (ow+1);
  float acc = supB[0];
  #pragma unroll 8
  for (int c = 0; c < 64; ++c) {
    float o = fe * base[(size_t)c*PLANE];
    base2[(size_t)(b*64 + c)*OPLANE + rem] = o;
    acc += o * supW[c];
  }
  supOut[idx] = acc;
}

// ===========================================================================
extern "C" void kernel_launch(void* const* d_in, const int* in_sizes, int n_in,
                              void* d_out, int out_size, void* d_ws, size_t ws_size,
                              hipStream_t stream) {
  (void)in_sizes; (void)n_in; (void)out_size; (void)ws_size;
  const float* fdm    = (const float*)d_in[0];
  const float* x2     = (const float*)d_in[3];
  const float* x3     = (const float*)d_in[4];
  const float* x4     = (const float*)d_in[5];
  const float* fuseW  = (const float*)d_in[6];
  const float* fuseB  = (const float*)d_in[7];
  const float* fuseG  = (const float*)d_in[8];
  const float* fuseBe = (const float*)d_in[9];
  const float* bcgW   = (const float*)d_in[10];
  const float* bcgB   = (const float*)d_in[11];
  const float* f0W    = (const float*)d_in[12];
  const float* f0B    = (const float*)d_in[13];
  const float* f0G    = (const float*)d_in[14];
  const float* f0Be   = (const float*)d_in[15];
  const float* supW   = (const float*)d_in[16];
  const float* supB   = (const float*)d_in[17];

  float* out   = (float*)d_out;
  float* o_fdm = out;                 // (4,1,300,300)
  float* o_bcg = out + 360000;        // (4,1,300,300)
  float* o_b2  = out + 720000;        // (4,64,300,300)
  float* o_sup = out + 23760000;      // (4,1,300,300)

  // workspace layout (floats); total ~41.8M floats = 167.3 MB
  float* ws = (float*)d_ws;
  float* xcat    = ws;                      // 192*91204 (aliases nl_raw below)
  float* nl_raw  = ws;                      // 64*90000 per-batch, reuses xcat
  size_t off = (size_t)192*PLANE;
  float* basePad = ws + off;  off += (size_t)4*64*PLANE;   // padded base, all b
  float* tmpH    = ws + off;  off += (size_t)64*HW*30;
  float* nf30    = ws + off;  off += (size_t)4*64*900;
  float* pooled  = ws + off;  off += 3600;
  float* nlbuf   = ws + off;  off += 3600;
  float* wtF     = ws + off;  off += (size_t)9*192*64;
  float* wtF0    = ws + off;  off += (size_t)9*64*64;
  float* stats   = ws + off;  off += 32;   // [0..7] fuse GN, [8..15] nl GN, [16..19] bcg sum

  init_kernel<<<(4*64*1204 + 255)/256, 256, 0, stream>>>(basePad, stats);
  wt_transform<<<(9*192*64)/256, 256, 0, stream>>>(fuseW, wtF, 192);
  wt_transform<<<(9*64*64)/256, 256, 0, stream>>>(f0W, wtF0, 64);

  for (int b = 0; b < 4; ++b) {
    float* bpB = basePad + (size_t)b*64*PLANE;
    resize_concat_1b<<<(192*PLANE)/256, 256, 0, stream>>>(x2, x3, x4, xcat, b);
    conv3x3_wmma<192><<<dim3(3,300), 256, 0, stream>>>(xcat, wtF, fuseB,
        bpB, PLANE, PW, PW + 1);
    gn_stats_1<<<512, 256, 0, stream>>>(bpB, PLANE, PW, PW + 1, stats + 2*b);
    gn_norm_relu_1<<<(64*OPLANE)/256, 256, 0, stream>>>(bpB, PLANE, PW, PW + 1,
        stats + 2*b, fuseG, fuseBe);
    conv3x3_wmma<64><<<dim3(3,300), 256, 0, stream>>>(bpB, wtF0, f0B,
        nl_raw, OPLANE, HW, 0);
    gn_stats_1<<<512, 256, 0, stream>>>(nl_raw, OPLANE, HW, 0, stats + 8 + 2*b);
    gn_norm_relu_1<<<(64*OPLANE)/256, 256, 0, stream>>>(nl_raw, OPLANE, HW, 0,
        stats + 8 + 2*b, f0G, f0Be);
    resize_w_down_1b<<<(64*HW*30)/256, 256, 0, stream>>>(nl_raw, tmpH);
    resize_h_down_1b<<<(64*900)/256, 256, 0, stream>>>(tmpH, nf30 + (size_t)b*64*900);
  }

  bcg_kernel<<<dim3(352,4), 256, 0, stream>>>(basePad, bcgW, bcgB, o_bcg, stats + 16);
  fdm_mask_kernel<<<(4*OPLANE + 255)/256, 256, 0, stream>>>(fdm, o_bcg, stats + 16, o_fdm);
  pool_kernel<<<(3600 + 255)/256, 256, 0, stream>>>(o_fdm, pooled);
  nl_einsum_wmma<<<dim3(57,4), 256, 0, stream>>>(nf30, pooled, nlbuf);
  final_kernel<<<(4*OPLANE + 255)/256, 256, 0, stream>>>(basePad, nlbuf, supW, supB,
                                                         o_b2, o_sup);
}